// DJ_supervised_41884521071058
// MI455X (gfx1250) — compile-verified
//
#include <hip/hip_runtime.h>
#include <hip/hip_bf16.h>

#define N_NODES 50000
#define EDGES   800000
#define F       256
#define NADJ    4
#define C_OUT   40
#define BN_EPS  1e-5f

typedef __attribute__((ext_vector_type(2))) float v2f;
typedef __attribute__((ext_vector_type(8))) float v8f;

// ---------------------------------------------------------------------------
// mask = softmax(att)  (4 values)
// ---------------------------------------------------------------------------
__global__ void k_mask(const float* __restrict__ att, float* __restrict__ mask) {
    if (threadIdx.x == 0 && blockIdx.x == 0) {
        float m = att[0];
        for (int i = 1; i < NADJ; ++i) m = fmaxf(m, att[i]);
        float e[NADJ]; float s = 0.f;
        for (int i = 0; i < NADJ; ++i) { e[i] = expf(att[i] - m); s += e[i]; }
        for (int i = 0; i < NADJ; ++i) mask[i] = e[i] / s;
    }
}

// ---------------------------------------------------------------------------
// zero fill, float4 (n must be a multiple of 4)
// ---------------------------------------------------------------------------
__global__ void k_zero4(float4* __restrict__ p, int n4) {
    int i = blockIdx.x * blockDim.x + threadIdx.x;
    if (i < n4) p[i] = make_float4(0.f, 0.f, 0.f, 0.f);
}

__global__ void k_zero(float* __restrict__ p, int n) {
    int i = blockIdx.x * blockDim.x + threadIdx.x;
    if (i < n) p[i] = 0.0f;
}

// ---------------------------------------------------------------------------
// degree counting: deg[a*N + dst] += 1 for every edge of every adjacency
// ---------------------------------------------------------------------------
__global__ void k_deg(const int* __restrict__ adj, float* __restrict__ deg) {
    int idx = blockIdx.x * blockDim.x + threadIdx.x;
    if (idx >= NADJ * EDGES) return;
    int a = idx / EDGES;
    int e = idx - a * EDGES;
    int d = adj[(size_t)a * 2 * EDGES + EDGES + e];
    unsafeAtomicAdd(&deg[(size_t)a * N_NODES + d], 1.0f);   // global_atomic_add_f32
}

// deg += 1 (self loop), dinv = rsqrt(deg)
__global__ void k_degfin(float* __restrict__ deg, float* __restrict__ dinv, int n) {
    int i = blockIdx.x * blockDim.x + threadIdx.x;
    if (i >= n) return;
    float d = deg[i] + 1.0f;
    deg[i]  = d;
    dinv[i] = rsqrtf(d);
}

// ---------------------------------------------------------------------------
// 256x256 transpose: Wt[n][k] = W[k][n]  (makes B fragments contiguous pairs)
// ---------------------------------------------------------------------------
__global__ void k_transpose256(const float* __restrict__ W, float* __restrict__ Wt) {
    int idx = blockIdx.x * blockDim.x + threadIdx.x;   // 65536 threads
    int k = idx >> 8, n = idx & 255;
    Wt[(size_t)n * F + k] = W[(size_t)k * F + n];
}

// ---------------------------------------------------------------------------
// WMMA fp32 GEMM:  C[nrows,256] = A[nrows,256] @ W[256,256], W given transposed.
// 256 threads = 8 waves. Each wave: 32 rows x 64 cols = 2 M-tiles x 4 N-tiles
// (8 v8f accumulators). Per K-step (K+=4): 2 A b64 loads + 4 B b64 loads feed
// 8 V_WMMA_F32_16X16X4_F32 ops. fp32 A/B fragment layout: lane&15 = row/col,
// lane>>4 selects the K pair {0,1}/{2,3} -> contiguous v2f per lane.
// ---------------------------------------------------------------------------
__global__ __launch_bounds__(256) void k_gemm256(const float* __restrict__ A,
                                                 const float* __restrict__ Wt,
                                                 float* __restrict__ Cmat,
                                                 int nrows) {
    const int lane = threadIdx.x & 31;
    const int wave = threadIdx.x >> 5;
    const int row0 = blockIdx.x * 64 + (wave >> 2) * 32;   // wave: rows row0..row0+31
    if (row0 >= nrows) return;
    const bool valid1 = (row0 + 16) < nrows;
    const int nbase = (wave & 3) * 64;
    const int lm    = lane & 15;
    const int kh    = (lane >> 4) << 1;      // 0 or 2

    const float* arow0 = A + (size_t)(row0 + lm) * F + kh;
    const float* arow1 = valid1 ? (arow0 + (size_t)16 * F) : arow0;  // branch-free guard
    const float* w0 = Wt + (size_t)(nbase + lm) * F + kh;        // tile 0 column row
    const float* w1 = w0 + (size_t)16 * F;
    const float* w2 = w0 + (size_t)32 * F;
    const float* w3 = w0 + (size_t)48 * F;

    v8f c00 = {}; v8f c01 = {}; v8f c02 = {}; v8f c03 = {};
    v8f c10 = {}; v8f c11 = {}; v8f c12 = {}; v8f c13 = {};

#pragma unroll 2
    for (int k = 0; k < F; k += 4) {
        v2f a0 = *(const v2f*)(arow0 + k);
        v2f a1 = *(const v2f*)(arow1 + k);
        v2f b0 = *(const v2f*)(w0 + k);
        v2f b1 = *(const v2f*)(w1 + k);
        v2f b2 = *(const v2f*)(w2 + k);
        v2f b3 = *(const v2f*)(w3 + k);
        c00 = __builtin_amdgcn_wmma_f32_16x16x4_f32(false, a0, false, b0, (short)0, c00, false, false);
        c10 = __builtin_amdgcn_wmma_f32_16x16x4_f32(false, a1, false, b0, (short)0, c10, false, false);
        c01 = __builtin_amdgcn_wmma_f32_16x16x4_f32(false, a0, false, b1, (short)0, c01, false, false);
        c11 = __builtin_amdgcn_wmma_f32_16x16x4_f32(false, a1, false, b1, (short)0, c11, false, false);
        c02 = __builtin_amdgcn_wmma_f32_16x16x4_f32(false, a0, false, b2, (short)0, c02, false, false);
        c12 = __builtin_amdgcn_wmma_f32_16x16x4_f32(false, a1, false, b2, (short)0, c12, false, false);
        c03 = __builtin_amdgcn_wmma_f32_16x16x4_f32(false, a0, false, b3, (short)0, c03, false, false);
        c13 = __builtin_amdgcn_wmma_f32_16x16x4_f32(false, a1, false, b3, (short)0, c13, false, false);
    }

    // C/D 16x16 layout: VGPR i -> row tile_row0 + (lane>=16 ? 8 : 0) + i, col lane&15
    const int rsub = (lane >> 4) << 3;
    float* out0 = Cmat + (size_t)(row0 + rsub) * F + nbase + lm;
#pragma unroll
    for (int i = 0; i < 8; ++i) {
        out0[(size_t)i * F + 0]  = c00[i];
        out0[(size_t)i * F + 16] = c01[i];
        out0[(size_t)i * F + 32] = c02[i];
        out0[(size_t)i * F + 48] = c03[i];
    }
    if (valid1) {
        float* out1 = out0 + (size_t)16 * F;
#pragma unroll
        for (int i = 0; i < 8; ++i) {
            out1[(size_t)i * F + 0]  = c10[i];
            out1[(size_t)i * F + 16] = c11[i];
            out1[(size_t)i * F + 32] = c12[i];
            out1[(size_t)i * F + 48] = c13[i];
        }
    }
}

// ---------------------------------------------------------------------------
// edge scatter: out[dst] += dinv[src]*dinv[dst] * h[src]
// 64 lanes per edge, float4 per lane (256 features). h/out are L2-resident.
// ---------------------------------------------------------------------------
__global__ __launch_bounds__(256) void k_scatter(const int* __restrict__ src,
                                                 const int* __restrict__ dst,
                                                 const float* __restrict__ dinv,
                                                 const float* __restrict__ h,
                                                 float* __restrict__ out) {
    const int e = blockIdx.x * 4 + (threadIdx.x >> 6);
    if (e >= EDGES) return;
    const int t = threadIdx.x & 63;
    const int s = src[e], d = dst[e];
    const float norm = dinv[s] * dinv[d];
    const float4 v = *(const float4*)(h + (size_t)s * F + t * 4);
    float* o = out + (size_t)d * F + t * 4;
    unsafeAtomicAdd(o + 0, norm * v.x);
    unsafeAtomicAdd(o + 1, norm * v.y);
    unsafeAtomicAdd(o + 2, norm * v.z);
    unsafeAtomicAdd(o + 3, norm * v.w);
}

// ---------------------------------------------------------------------------
// epilogue: buf = relu( BN?( buf + h/deg + bias ) ) * mask[mask_idx]
// ---------------------------------------------------------------------------
__global__ __launch_bounds__(256) void k_finalize(float* __restrict__ buf,
                                                  const float* __restrict__ h,
                                                  const float* __restrict__ deg,
                                                  const float* __restrict__ bias,
                                                  int use_bn,
                                                  const float* __restrict__ g,
                                                  const float* __restrict__ be,
                                                  const float* __restrict__ mu,
                                                  const float* __restrict__ var,
                                                  const float* __restrict__ mask,
                                                  int mask_idx) {
    int idx = blockIdx.x * blockDim.x + threadIdx.x;
    if (idx >= N_NODES * F) return;
    int n = idx >> 8, f = idx & 255;
    float val = buf[idx] + h[idx] / deg[n] + bias[f];
    if (use_bn)
        val = (val - mu[f]) * rsqrtf(var[f] + BN_EPS) * g[f] + be[f];
    val = fmaxf(val, 0.0f);
    if (mask_idx >= 0) val *= mask[mask_idx];
    buf[idx] = val;
}

// ---------------------------------------------------------------------------
// classifier: logits initialized to bias; each branch adds Z_b @ Wcls_b
// ---------------------------------------------------------------------------
__global__ void k_logits_init(const float* __restrict__ bc, float* __restrict__ logits) {
    int idx = blockIdx.x * blockDim.x + threadIdx.x;
    if (idx >= N_NODES * C_OUT) return;
    logits[idx] = bc[idx % C_OUT];
}

__global__ __launch_bounds__(256) void k_logits_part(const float* __restrict__ Z,
                                                     const float* __restrict__ Wc,
                                                     float* __restrict__ logits) {
    int idx = blockIdx.x * blockDim.x + threadIdx.x;
    if (idx >= N_NODES * C_OUT) return;
    int n = idx / C_OUT, c = idx - n * C_OUT;
    const float* z = Z + (size_t)n * F;
    const float* w = Wc + c;
    float acc = 0.0f;
#pragma unroll 8
    for (int k = 0; k < F; ++k) acc += z[k] * w[(size_t)k * C_OUT];
    logits[idx] += acc;
}

// log_softmax per node; also writes the trailing scalar output (0.0)
__global__ void k_lsm(const float* __restrict__ logits, float* __restrict__ out) {
    int n = blockIdx.x * blockDim.x + threadIdx.x;
    if (n == 0 && blockIdx.x == 0) out[(size_t)N_NODES * C_OUT] = 0.0f;
    if (n >= N_NODES) return;
    const float* r = logits + (size_t)n * C_OUT;
    float m = r[0];
    for (int c = 1; c < C_OUT; ++c) m = fmaxf(m, r[c]);
    float s = 0.f;
    for (int c = 0; c < C_OUT; ++c) s += expf(r[c] - m);
    float lse = m + logf(s);
    float* o = out + (size_t)n * C_OUT;
    for (int c = 0; c < C_OUT; ++c) o[c] = r[c] - lse;
}

// ---------------------------------------------------------------------------
extern "C" void kernel_launch(void* const* d_in, const int* in_sizes, int n_in,
                              void* d_out, int out_size, void* d_ws, size_t ws_size,
                              hipStream_t stream) {
    const float* x        = (const float*)d_in[0];
    const int*   adj      = (const int*)  d_in[1];
    const float* W_convs  = (const float*)d_in[2];
    const float* b_convs  = (const float*)d_in[3];
    const float* W_extra  = (const float*)d_in[4];
    const float* b_extra  = (const float*)d_in[5];
    const float* bn1_g    = (const float*)d_in[6];
    const float* bn1_b    = (const float*)d_in[7];
    const float* bn1_m    = (const float*)d_in[8];
    const float* bn1_v    = (const float*)d_in[9];
    const float* W_extra2 = (const float*)d_in[10];
    const float* b_extra2 = (const float*)d_in[11];
    const float* bn2_g    = (const float*)d_in[12];
    const float* bn2_b    = (const float*)d_in[13];
    const float* bn2_m    = (const float*)d_in[14];
    const float* bn2_v    = (const float*)d_in[15];
    const float* att      = (const float*)d_in[16];
    const float* W_cls    = (const float*)d_in[17];
    const float* b_cls    = (const float*)d_in[18];
    float* out = (float*)d_out;

    const size_t NF = (size_t)N_NODES * F;
    float* ws     = (float*)d_ws;
    float* mask   = ws;                                  // 4
    float* deg    = mask + 4;                            // 4*N
    float* dinv   = deg  + (size_t)NADJ * N_NODES;       // 4*N
    float* tmp    = dinv + (size_t)NADJ * N_NODES;       // N*F  (GEMM output)
    float* B      = tmp  + NF;                           // N*F  (branch buffer)
    float* logits = B    + NF;                           // N*C_OUT
    float* Wt     = logits + (size_t)N_NODES * C_OUT;    // 256*256 transposed weight

    const int TB = 256;
    const int gemm_grid = (N_NODES + 63) / 64;
    const int nf4_grid  = (int)((NF / 4 + TB - 1) / TB);
    const int sc_grid   = EDGES / 4;
    const int lg_grid   = (N_NODES * C_OUT + TB - 1) / TB;
    const int tr_grid   = (F * F) / TB;

    // softmax(att)
    hipLaunchKernelGGL(k_mask, dim3(1), dim3(64), 0, stream, att, mask);

    // degrees + dinv for all 4 adjacencies
    hipLaunchKernelGGL(k_zero, dim3((NADJ * N_NODES + TB - 1) / TB), dim3(TB), 0, stream,
                       deg, NADJ * N_NODES);
    hipLaunchKernelGGL(k_deg, dim3((NADJ * EDGES + TB - 1) / TB), dim3(TB), 0, stream,
                       adj, deg);
    hipLaunchKernelGGL(k_degfin, dim3((NADJ * N_NODES + TB - 1) / TB), dim3(TB), 0, stream,
                       deg, dinv, NADJ * N_NODES);

    // logits = b_cls broadcast
    hipLaunchKernelGGL(k_logits_init, dim3(lg_grid), dim3(TB), 0, stream, b_cls, logits);

    // ---- jump branches i = 0..2, adjacency i+1 ----
    for (int i = 0; i < 3; ++i) {
        const int a = i + 1;
        const int* srcp = adj + (size_t)a * 2 * EDGES;
        const int* dstp = srcp + EDGES;
        hipLaunchKernelGGL(k_transpose256, dim3(tr_grid), dim3(TB), 0, stream,
                           W_convs + (size_t)i * F * F, Wt);
        hipLaunchKernelGGL(k_gemm256, dim3(gemm_grid), dim3(TB), 0, stream,
                           x, Wt, tmp, N_NODES);
        hipLaunchKernelGGL(k_zero4, dim3(nf4_grid), dim3(TB), 0, stream,
                           (float4*)B, (int)(NF / 4));
        hipLaunchKernelGGL(k_scatter, dim3(sc_grid), dim3(TB), 0, stream,
                           srcp, dstp, dinv + (size_t)a * N_NODES, tmp, B);
        hipLaunchKernelGGL(k_finalize, dim3((int)((NF + TB - 1) / TB)), dim3(TB), 0, stream,
                           B, tmp, deg + (size_t)a * N_NODES, b_convs + (size_t)i * F,
                           0, (const float*)nullptr, (const float*)nullptr,
                           (const float*)nullptr, (const float*)nullptr, mask, i);
        hipLaunchKernelGGL(k_logits_part, dim3(lg_grid), dim3(TB), 0, stream,
                           B, W_cls + (size_t)i * F * C_OUT, logits);
    }

    // ---- extra branch, adjacency 0, two GCN layers ----
    {
        const int* srcp = adj;
        const int* dstp = adj + EDGES;
        // layer 1: relu(bn1(agg(x@W_extra) + b_extra))
        hipLaunchKernelGGL(k_transpose256, dim3(tr_grid), dim3(TB), 0, stream, W_extra, Wt);
        hipLaunchKernelGGL(k_gemm256, dim3(gemm_grid), dim3(TB), 0, stream,
                           x, Wt, tmp, N_NODES);
        hipLaunchKernelGGL(k_zero4, dim3(nf4_grid), dim3(TB), 0, stream,
                           (float4*)B, (int)(NF / 4));
        hipLaunchKernelGGL(k_scatter, dim3(sc_grid), dim3(TB), 0, stream,
                           srcp, dstp, dinv, tmp, B);
        hipLaunchKernelGGL(k_finalize, dim3((int)((NF + TB - 1) / TB)), dim3(TB), 0, stream,
                           B, tmp, deg, b_extra, 1, bn1_g, bn1_b, bn1_m, bn1_v, mask, -1);
        // layer 2: relu(bn2(agg(e1@W_extra2) + b_extra2)) * mask[3]
        hipLaunchKernelGGL(k_transpose256, dim3(tr_grid), dim3(TB), 0, stream, W_extra2, Wt);
        hipLaunchKernelGGL(k_gemm256, dim3(gemm_grid), dim3(TB), 0, stream,
                           B, Wt, tmp, N_NODES);
        hipLaunchKernelGGL(k_zero4, dim3(nf4_grid), dim3(TB), 0, stream,
                           (float4*)B, (int)(NF / 4));
        hipLaunchKernelGGL(k_scatter, dim3(sc_grid), dim3(TB), 0, stream,
                           srcp, dstp, dinv, tmp, B);
        hipLaunchKernelGGL(k_finalize, dim3((int)((NF + TB - 1) / TB)), dim3(TB), 0, stream,
                           B, tmp, deg, b_extra2, 1, bn2_g, bn2_b, bn2_m, bn2_v, mask, 3);
        hipLaunchKernelGGL(k_logits_part, dim3(lg_grid), dim3(TB), 0, stream,
                           B, W_cls + (size_t)3 * F * C_OUT, logits);
    }

    // log_softmax + trailing scalar
    hipLaunchKernelGGL(k_lsm, dim3((N_NODES + TB - 1) / TB), dim3(TB), 0, stream,
                       logits, out);
}